// SphericalHarmonicsRel_69561290326721
// MI455X (gfx1250) — compile-verified
//
#include <hip/hip_runtime.h>
#include <cstdint>

// ---------- vector types with relaxed (4B) alignment so we can issue wide
// ---------- coalesced b128/b64 stores at any dword-aligned address ----------
typedef float f4v __attribute__((ext_vector_type(4)));
typedef float f2v __attribute__((ext_vector_type(2)));
typedef f4v __attribute__((aligned(4))) f4u;
typedef f2v __attribute__((aligned(4))) f2u;

struct cpx { float r, i; };
__device__ __forceinline__ cpx cmul(cpx a, cpx b) {
    return { fmaf(a.r, b.r, -a.i * b.i), fmaf(a.r, b.i, a.i * b.r) };
}
__device__ __forceinline__ cpx csc(float w, cpx a) { return { w * a.r, w * a.i }; }

__device__ __forceinline__ void st4(float* p, float a, float b, float c, float d) {
    f4v v = { a, b, c, d };
    __builtin_nontemporal_store(v, reinterpret_cast<f4u*>(p));   // global_store_b128 th:NT
}
__device__ __forceinline__ void st2(float* p, float a, float b) {
    f2v v = { a, b };
    __builtin_nontemporal_store(v, reinterpret_cast<f2u*>(p));   // global_store_b64 th:NT
}

#define NB   8
#define NPTS 512

__global__ __launch_bounds__(256)
void sh_rel_kernel(const float* __restrict__ pos1,
                   const float* __restrict__ pos2,
                   float* __restrict__ out)
{
    constexpr size_t M = (size_t)NB * NPTS * NPTS;      // 2,097,152 pairs
    const unsigned tid = threadIdx.x;
    const size_t   gid = (size_t)blockIdx.x * 256u + tid;
    const unsigned b   = (unsigned)(gid >> 18);          // 512*512 = 2^18
    const unsigned i   = ((unsigned)gid >> 9) & 511u;    // uniform per block
    const unsigned j0  = (blockIdx.x & 1u) * 256u;       // block's first j

    // ---- CDNA5 async DMA: stage this block's 256 pos2 points (3 KB) into LDS.
    // Each wave fetches exactly the 96 dwords its own lanes consume, so a
    // wave-local s_wait_asynccnt suffices (no workgroup barrier).
    // GV addressing form: 64-bit per-lane address in a VGPR pair, saddr = off.
    __shared__ float sP2[256 * 3];
    {
        const float* gsrc = pos2 + ((size_t)b * NPTS + j0) * 3;
        const unsigned wbase   = (tid & ~31u) * 3u;              // wave's dword base
        const unsigned lane    = tid & 31u;
        const unsigned ldsBase = (unsigned)(uintptr_t)(&sP2[0]); // flat low 32b == LDS offset
        #pragma unroll
        for (int k = 0; k < 3; ++k) {
            unsigned dw    = wbase + lane + (unsigned)k * 32u;
            unsigned voff  = dw * 4u;
            unsigned laddr = ldsBase + voff;
            unsigned long long gaddr =
                (unsigned long long)(uintptr_t)gsrc + (unsigned long long)voff;
            asm volatile("global_load_async_to_lds_b32 %0, %1, off"
                         :: "v"(laddr), "v"(gaddr) : "memory");
        }
        asm volatile("s_wait_asynccnt 0" ::: "memory");
    }

    // ---- uniform pos1 row (hits scalar cache; prefetch the line) ----
    const float* p1 = pos1 + ((size_t)b * NPTS + i) * 3;
    __builtin_prefetch(p1, 0, 3);                         // global_prefetch_b8
    const float p1x = p1[0], p1y = p1[1], p1z = p1[2];

    const float p2x = sP2[tid * 3 + 0];
    const float p2y = sP2[tid * 3 + 1];
    const float p2z = sP2[tid * 3 + 2];

    const float rx = p1x - p2x, ry = p1y - p2y, rz = p1z - p2z;
    const float n2 = fmaf(rx, rx, fmaf(ry, ry, rz * rz));
    const float n  = sqrtf(n2);
    const float inv = (n > 0.f) ? (1.0f / n) : 0.0f;
    const float x = rx * inv, y = ry * inv, z = rz * inv;

    // ---- compile-time Clebsch-Gordan / normalization constants ----
    const float C0  = 0.28209479177387814f;  // sqrt(1/(4pi))
    const float C1  = 0.48860251190291992f;  // sqrt(3/(4pi))
    const float IS2 = 0.70710678118654752f;  // 1/sqrt(2)
    const float S2  = 3.2360432f;            // sqrt(10*pi/3)
    const float S3  = 3.1263161f;            // (2/3)*sqrt(7*pi)
    const float SQ2 = 1.41421356237309505f;
    const float W2A = 0.40824829046386302f;  // 1/sqrt(6)
    const float W2B = 0.81649658092772603f;  // sqrt(2/3)
    const float W3A = 0.81649658092772603f;  // sqrt(2/3)
    const float W3B = 0.57735026918962576f;  // sqrt(1/3)
    const float W3C = 0.63245553203367587f;  // sqrt(2/5)
    const float W3D = 0.73029674334022143f;  // sqrt(8/15)
    const float W3E = 0.25819888974716110f;  // sqrt(1/15)
    const float W3F = 0.44721359549995794f;  // sqrt(1/5)
    const float W3G = 0.77459666924148338f;  // sqrt(3/5)

    // ---- l = 1 (includes C1 factor, matches reference's psi1) ----
    const cpx bm = {  x * IS2 * C1, -y * IS2 * C1 };
    const cpx b0 = {  z * C1,        0.0f         };
    const cpx bp = { -x * IS2 * C1, -y * IS2 * C1 };

    // ---- l = 2: q[m] = S2 * sum_m1 W2[m1][m] * psi1[m-m1] * psi1[m1] ----
    const cpx q0 = csc(S2, cmul(bm, bm));
    const cpx q1 = csc(S2 * SQ2, cmul(bm, b0));
    cpx q2; {
        cpx u = cmul(bm, bp), v = cmul(b0, b0);
        q2 = { S2 * fmaf(2.f * W2A, u.r, W2B * v.r),
               S2 * fmaf(2.f * W2A, u.i, W2B * v.i) };
    }
    const cpx q3 = csc(S2 * SQ2, cmul(bp, b0));
    const cpx q4 = csc(S2, cmul(bp, bp));

    // ---- l = 3: t[m] = S3 * sum_m1 W3[m1][m] * psi2[m-m1] * psi1[m1] ----
    const cpx t0 = csc(S3, cmul(q0, bm));
    cpx t1; { cpx u = cmul(q1, bm), v = cmul(q0, b0);
        t1 = { S3 * fmaf(W3A, u.r, W3B * v.r), S3 * fmaf(W3A, u.i, W3B * v.i) }; }
    cpx t2; { cpx u = cmul(q2, bm), v = cmul(q1, b0), w = cmul(q0, bp);
        t2 = { S3 * fmaf(W3C, u.r, fmaf(W3D, v.r, W3E * w.r)),
               S3 * fmaf(W3C, u.i, fmaf(W3D, v.i, W3E * w.i)) }; }
    cpx t3; { cpx u = cmul(q3, bm), v = cmul(q2, b0), w = cmul(q1, bp);
        t3 = { S3 * fmaf(W3F, u.r, fmaf(W3G, v.r, W3F * w.r)),
               S3 * fmaf(W3F, u.i, fmaf(W3G, v.i, W3F * w.i)) }; }
    cpx t4; { cpx u = cmul(q4, bm), v = cmul(q3, b0), w = cmul(q2, bp);
        t4 = { S3 * fmaf(W3E, u.r, fmaf(W3D, v.r, W3C * w.r)),
               S3 * fmaf(W3E, u.i, fmaf(W3D, v.i, W3C * w.i)) }; }
    cpx t5; { cpx u = cmul(q4, b0), v = cmul(q3, bp);
        t5 = { S3 * fmaf(W3B, u.r, W3A * v.r), S3 * fmaf(W3B, u.i, W3A * v.i) }; }
    const cpx t6 = csc(S3, cmul(q4, bp));

    // ---- streaming NT stores: outputs concatenated (psi0,psi1,psi2,psi3,norms)
    float* o0 = out;                 // M*2
    float* o1 = out + 2  * M;        // M*6
    float* o2 = out + 8  * M;        // M*10
    float* o3 = out + 18 * M;        // M*14
    float* on = out + 32 * M;        // M

    st2(o0 + gid * 2, C0, 0.0f);

    st4(o1 + gid * 6,     bm.r, bm.i, b0.r, b0.i);
    st2(o1 + gid * 6 + 4, bp.r, bp.i);

    st4(o2 + gid * 10,     q0.r, q0.i, q1.r, q1.i);
    st4(o2 + gid * 10 + 4, q2.r, q2.i, q3.r, q3.i);
    st2(o2 + gid * 10 + 8, q4.r, q4.i);

    st4(o3 + gid * 14,      t0.r, t0.i, t1.r, t1.i);
    st4(o3 + gid * 14 + 4,  t2.r, t2.i, t3.r, t3.i);
    st4(o3 + gid * 14 + 8,  t4.r, t4.i, t5.r, t5.i);
    st2(o3 + gid * 14 + 12, t6.r, t6.i);

    __builtin_nontemporal_store(n, on + gid);
}

extern "C" void kernel_launch(void* const* d_in, const int* in_sizes, int n_in,
                              void* d_out, int out_size, void* d_ws, size_t ws_size,
                              hipStream_t stream) {
    (void)in_sizes; (void)n_in; (void)out_size; (void)d_ws; (void)ws_size;
    const float* pos1 = (const float*)d_in[0];
    const float* pos2 = (const float*)d_in[1];
    float* out = (float*)d_out;
    constexpr size_t M = (size_t)NB * NPTS * NPTS;
    dim3 grid((unsigned)(M / 256)), block(256);
    hipLaunchKernelGGL(sh_rel_kernel, grid, block, 0, stream, pos1, pos2, out);
}